// GCNLayer_12730283065988
// MI455X (gfx1250) — compile-verified
//
#include <hip/hip_runtime.h>

typedef __attribute__((ext_vector_type(2))) float v2f;
typedef __attribute__((ext_vector_type(8))) float v8f;
typedef int v4i __attribute__((vector_size(16)));

#define DFEAT 128
#define LDA   130   // padded LDS row stride (floats) to avoid bank conflicts

// CDNA5 async global->LDS staging if the toolchain exposes it; otherwise
// fall back to a plain synchronous copy (still compiles to the same WMMA mix).
#if defined(__has_builtin)
#if __has_builtin(__builtin_amdgcn_global_load_async_to_lds_b128) && \
    __has_builtin(__builtin_amdgcn_s_wait_asynccnt)
#define USE_ASYNC_LDS 1
#endif
#endif

// ---------------------------------------------------------------------------
// Phase 2: edge scatter.  One wave (32 lanes) per edge; 4 floats per lane.
// src/dst are wave-uniform -> scalar loads.  Hardware f32 atomics into the
// L2-resident accumulator (d_out reused as `summed`).
// ---------------------------------------------------------------------------
__global__ __launch_bounds__(256) void gcn_edge_scatter(
    const int* __restrict__ src, const int* __restrict__ dst,
    const float* __restrict__ h, const float* __restrict__ r,
    float* __restrict__ summed, float* __restrict__ counts, int E)
{
    int t    = blockIdx.x * 256 + threadIdx.x;
    int e    = t >> 5;
    if (e >= E) return;
    int lane = t & 31;
    int s    = src[e];
    int d    = dst[e];
    int col  = lane * 4;

    const float4 hv = *reinterpret_cast<const float4*>(h + (size_t)s * DFEAT + col);
    const float4 rv = *reinterpret_cast<const float4*>(r + (size_t)e * DFEAT + col);

    float* outp = summed + (size_t)d * DFEAT + col;
    unsafeAtomicAdd(outp + 0, hv.x + rv.x);
    unsafeAtomicAdd(outp + 1, hv.y + rv.y);
    unsafeAtomicAdd(outp + 2, hv.z + rv.z);
    unsafeAtomicAdd(outp + 3, hv.w + rv.w);
    if (lane == 0) unsafeAtomicAdd(counts + d, 1.0f);
}

// ---------------------------------------------------------------------------
// Phase 3: out = (summed @ W^T) * s_row + b  with  s_row = 1/max(count,1),
// via V_WMMA_F32_16X16X4_F32.  One block per 16-node tile; wave t (of 8)
// computes output-feature columns [16t, 16t+16).  In-place on d_out (tile
// fully staged into LDS before any store).  The mean scale is applied to the
// accumulator in the epilogue (valid since it is constant per output row).
//
// WMMA f32 16x16x4 lane layouts (ISA 7.12.2):
//   A (16x4):  lane L holds A[L%16][ (L/16)*2 + i ]  for i in {0,1}
//   B (4x16):  lane L holds B[ (L/16)*2 + i ][ L%16 ]
//   C/D:       VGPR v of lane L holds D[ v + 8*(L/16) ][ L%16 ]
// ---------------------------------------------------------------------------
__global__ __launch_bounds__(256) void gcn_gemm_wmma(
    const float* __restrict__ summed, const float* __restrict__ counts,
    const float* __restrict__ W, const float* __restrict__ bias,
    float* __restrict__ out, int N, int ntiles)
{
    __shared__ __align__(16) float ldsA[16 * LDA];
    __shared__ float ldsS[16];

    const int tid  = threadIdx.x;
    const int wave = tid >> 5;     // 0..7 : output feature tile
    const int lane = tid & 31;
    const int lo   = lane & 15;
    const int hi   = lane >> 4;    // 0 or 1

    // Preload this wave's W^T fragments: B[k][j] = W[j][k], j = wave*16+lo.
    // Per k-step ks: k = 4*ks + 2*hi + i  -> two consecutive floats of row j.
    v2f bw[32];
    const float* wrow = W + (size_t)(wave * 16 + lo) * DFEAT + hi * 2;
#pragma unroll
    for (int ks = 0; ks < 32; ++ks)
        bw[ks] = *reinterpret_cast<const v2f*>(wrow + ks * 4);

    const float bval = bias[wave * 16 + lo];

    for (int tile = blockIdx.x; tile < ntiles; tile += gridDim.x) {
        const int n0 = tile * 16;

        // Stage the raw 16x128 A tile into LDS (no scaling here).
        {
            int row = tid >> 4;          // 0..15
            int col = (tid & 15) * 8;    // 0..120 (32 bytes per thread)
            int rr  = n0 + row; if (rr > N - 1) rr = N - 1;
            const float* sp = summed + (size_t)rr * DFEAT + col;
            float*       dp = &ldsA[row * LDA + col];
#if defined(USE_ASYNC_LDS)
            // Builtin signature (from hipcc diagnostic): arg0 is a GLOBAL
            // (AS1) v4i* source, arg1 the LDS (AS3) v4i* destination.
            typedef __attribute__((address_space(1))) v4i gv4i;
            typedef __attribute__((address_space(3))) v4i lv4i;
            __builtin_amdgcn_global_load_async_to_lds_b128(
                (gv4i*)(sp),     (lv4i*)(dp),     0, 0);
            __builtin_amdgcn_global_load_async_to_lds_b128(
                (gv4i*)(sp + 4), (lv4i*)(dp + 4), 0, 0);
#else
            *reinterpret_cast<float4*>(dp)     = *reinterpret_cast<const float4*>(sp);
            *reinterpret_cast<float4*>(dp + 4) = *reinterpret_cast<const float4*>(sp + 4);
#endif
        }
        // Per-row mean scale, computed once per tile.
        if (tid < 16) {
            int rr = n0 + tid; if (rr > N - 1) rr = N - 1;
            ldsS[tid] = 1.0f / fmaxf(counts[rr], 1.0f);
        }
#if defined(USE_ASYNC_LDS)
        __builtin_amdgcn_s_wait_asynccnt(0);   // drain this wave's async copies
#endif
        __syncthreads();                       // publish LDS across all waves

        v8f acc = {};
        const float* arow = &ldsA[lo * LDA + hi * 2];
#pragma unroll
        for (int ks = 0; ks < 32; ++ks) {
            v2f a = *reinterpret_cast<const v2f*>(arow + ks * 4);
            acc = __builtin_amdgcn_wmma_f32_16x16x4_f32(
                /*neg_a=*/false, a, /*neg_b=*/false, bw[ks],
                /*c_mod=*/(short)0, acc, /*reuse_a=*/false, /*reuse_b=*/false);
        }

        // Epilogue: per-row mean scale + bias.  In-place safe: tile staged.
#pragma unroll
        for (int v = 0; v < 8; ++v) {
            int   row = n0 + v + hi * 8;
            float sv  = ldsS[v + hi * 8];
            if (row < N)
                out[(size_t)row * DFEAT + wave * 16 + lo] = acc[v] * sv + bval;
        }
        __syncthreads();   // protect ldsA/ldsS before next iteration
    }
}

// ---------------------------------------------------------------------------
extern "C" void kernel_launch(void* const* d_in, const int* in_sizes, int n_in,
                              void* d_out, int out_size, void* d_ws, size_t ws_size,
                              hipStream_t stream) {
    const int*   src = (const int*)d_in[0];
    const int*   dst = (const int*)d_in[1];
    const float* h   = (const float*)d_in[2];
    const float* r   = (const float*)d_in[3];
    const float* W   = (const float*)d_in[4];
    const float* b   = (const float*)d_in[5];
    float*       out = (float*)d_out;

    const int E = in_sizes[0];
    const int N = out_size / DFEAT;          // 50000

    float* counts = (float*)d_ws;            // N floats of scratch

    // Zero accumulator (d_out doubles as `summed`) and counts.
    (void)hipMemsetAsync(out,    0, (size_t)out_size * sizeof(float), stream);
    (void)hipMemsetAsync(counts, 0, (size_t)N * sizeof(float), stream);

    // Edge scatter: one wave per edge -> E/8 blocks of 256 threads.
    int blocks_e = (E + 7) / 8;
    gcn_edge_scatter<<<blocks_e, 256, 0, stream>>>(src, dst, h, r, out, counts, E);

    // WMMA GEMM: one block per 16-node tile.
    int ntiles = (N + 15) / 16;              // 3125
    gcn_gemm_wmma<<<ntiles, 256, 0, stream>>>(out, counts, W, b, out, N, ntiles);
}